// TokenLabelHead_56848187129955
// MI455X (gfx1250) — compile-verified
//
#include <hip/hip_runtime.h>

// Dims from the reference
#define B_   16
#define S_   2048
#define DIN  1024
#define H_   512
#define L_   256
#define DEMB 256
#define ROWS (B_ * S_)   // 32768

typedef __attribute__((ext_vector_type(2))) float v2f;
typedef __attribute__((ext_vector_type(8))) float v8f;

// ws layout (floats):
//   [0    .. 1023] v_tok  = Wt^T @ wc_tok              (DIN)
//   [1024 .. 1279] v_lab  = Wl^T @ wc_lab              (DEMB)
//   [1280]         scalar = bt.wc_tok + bl.wc_lab + bc
//   [1296 .. 1551] lab_final[l] = label_emb[l,:].v_lab + scalar
#define WS_VTOK 0
#define WS_VLAB 1024
#define WS_SCAL 1280
#define WS_LABF 1296

// ---------------------------------------------------------------------------
// Kernel 1: fold the big weight matrices against Wc.
// blocks 0..3: v_tok (coalesced: thread d reads Wt[h*DIN + d])
// block 4:     v_lab
// block 5:     scalar (one wave)
// ---------------------------------------------------------------------------
__global__ __launch_bounds__(256) void tlh_prep_kernel(
    const float* __restrict__ Wt, const float* __restrict__ bt,
    const float* __restrict__ Wl, const float* __restrict__ bl,
    const float* __restrict__ Wc, const float* __restrict__ bc,
    float* __restrict__ ws) {
  const int b = blockIdx.x;
  const int tid = threadIdx.x;
  if (b < 4) {
    const int d = b * 256 + tid;            // 0..1023
    float acc = 0.f;
    for (int h = 0; h < H_; ++h)
      acc = fmaf(Wt[(size_t)h * DIN + d], Wc[h], acc);
    ws[WS_VTOK + d] = acc;
  } else if (b == 4) {
    const int e = tid;                      // 0..255
    float acc = 0.f;
    for (int h = 0; h < H_; ++h)
      acc = fmaf(Wl[(size_t)h * DEMB + e], Wc[H_ + h], acc);
    ws[WS_VLAB + e] = acc;
  } else {
    if (tid < 32) {
      float acc = 0.f;
      for (int h = tid; h < H_; h += 32)
        acc += bt[h] * Wc[h] + bl[h] * Wc[H_ + h];
      #pragma unroll
      for (int off = 16; off > 0; off >>= 1) acc += __shfl_xor(acc, off, 32);
      if (tid == 0) ws[WS_SCAL] = acc + bc[0];
    }
  }
}

// ---------------------------------------------------------------------------
// Kernel 2: lab_final[l] = label_emb[l,:] . v_lab + scalar   (256 outputs)
// ---------------------------------------------------------------------------
__global__ __launch_bounds__(256) void tlh_lab_kernel(
    const float* __restrict__ label_emb, float* __restrict__ ws) {
  __shared__ float vle[DEMB];
  const int l = threadIdx.x;
  vle[l] = ws[WS_VLAB + l];
  __syncthreads();
  const float s = ws[WS_SCAL];
  float acc = 0.f;
  #pragma unroll 4
  for (int e = 0; e < DEMB; ++e)
    acc = fmaf(label_emb[(size_t)l * DEMB + e], vle[e], acc);
  ws[WS_LABF + l] = acc + s;
}

// ---------------------------------------------------------------------------
// Kernel 3 (bandwidth kernel): one wave per row.
//   tok = seq[row,:] . v_tok   (coalesced float4 loads; v_tok in registers)
//   cross-lane reduction via V_WMMA_F32_16X16X4_F32 (B = ones):
//     lane m holds A[m][0:1], lane m+16 holds A[m][2:3]  -> D[m][n] = row sum
//   out[row, l] = tok + lab_final[l]   (2 coalesced float4 stores per lane)
// ---------------------------------------------------------------------------
__global__ __launch_bounds__(256) void tlh_main_kernel(
    const float* __restrict__ seq, const float* __restrict__ ws,
    float* __restrict__ out, int totalRows) {
  const int lane  = threadIdx.x & 31;
  const int wave  = threadIdx.x >> 5;
  const int wgid  = blockIdx.x * 8 + wave;     // global wave id
  const int nwave = gridDim.x * 8;

  const float4* __restrict__ vtok = (const float4*)(ws + WS_VTOK); // 256 f4
  const float4* __restrict__ labf = (const float4*)(ws + WS_LABF); // 64 f4

  // Register-resident v_tok slice: lane i holds v_tok[j*128 + 4i .. +3]
  float4 vt[8];
  #pragma unroll
  for (int j = 0; j < 8; ++j) vt[j] = vtok[j * 32 + lane];
  const float4 labA = labf[lane];
  const float4 labB = labf[32 + lane];

  const v2f ones = {1.0f, 1.0f};

  for (int row = wgid; row < totalRows; row += nwave) {
    const float4* __restrict__ r = (const float4*)(seq + (size_t)row * DIN);
    float4 acc = {0.f, 0.f, 0.f, 0.f};
    #pragma unroll
    for (int j = 0; j < 8; ++j) {
      const float4 x = r[j * 32 + lane];     // global_load_b128, coalesced
      acc.x = fmaf(x.x, vt[j].x, acc.x);
      acc.y = fmaf(x.y, vt[j].y, acc.y);
      acc.z = fmaf(x.z, vt[j].z, acc.z);
      acc.w = fmaf(x.w, vt[j].w, acc.w);
    }

    // WMMA reduction of the 32 lane partials (2 per lane -> full 16x4 A).
    v2f a;
    a.x = acc.x + acc.y;
    a.y = acc.z + acc.w;
    v8f c = {};
    c = __builtin_amdgcn_wmma_f32_16x16x4_f32(
        /*neg_a=*/false, a, /*neg_b=*/false, ones,
        /*c_mod=*/(short)0, c, /*reuse_a=*/false, /*reuse_b=*/false);
    float s = ((c[0] + c[1]) + (c[2] + c[3])) + ((c[4] + c[5]) + (c[6] + c[7]));
    const float tok = s + __shfl_xor(s, 16, 32);   // pair lanes n / n+16

    float4 oA, oB;
    oA.x = labA.x + tok; oA.y = labA.y + tok;
    oA.z = labA.z + tok; oA.w = labA.w + tok;
    oB.x = labB.x + tok; oB.y = labB.y + tok;
    oB.z = labB.z + tok; oB.w = labB.w + tok;

    float4* __restrict__ o = (float4*)(out + (size_t)row * L_);
    o[lane]      = oA;                       // global_store_b128, coalesced
    o[32 + lane] = oB;
  }
}

// ---------------------------------------------------------------------------
extern "C" void kernel_launch(void* const* d_in, const int* in_sizes, int n_in,
                              void* d_out, int out_size, void* d_ws, size_t ws_size,
                              hipStream_t stream) {
  const float* seq = (const float*)d_in[0];   // (16,2048,1024)
  const float* lab = (const float*)d_in[1];   // (256,256)
  const float* Wt  = (const float*)d_in[2];   // (512,1024)
  const float* bt  = (const float*)d_in[3];   // (512)
  const float* Wl  = (const float*)d_in[4];   // (512,256)
  const float* bl  = (const float*)d_in[5];   // (512)
  const float* Wc  = (const float*)d_in[6];   // (1,1024)
  const float* bc  = (const float*)d_in[7];   // (1)
  float* out = (float*)d_out;
  float* ws  = (float*)d_ws;

  tlh_prep_kernel<<<6, 256, 0, stream>>>(Wt, bt, Wl, bl, Wc, bc, ws);
  tlh_lab_kernel<<<1, 256, 0, stream>>>(lab, ws);
  tlh_main_kernel<<<2048, 256, 0, stream>>>(seq, ws, out, ROWS);
}